// GraphConv_26800595927296
// MI455X (gfx1250) — compile-verified
//
#include <hip/hip_runtime.h>
#include <math.h>
#include <limits.h>

// ---------------------------------------------------------------------------
// GraphConv (GAT-flavored) for MI455X / gfx1250, wave32.
//   h      = (feat * out_deg^-1/2) @ W          <- V_WMMA_F32_16X16X4_F32
//   e      = leaky_relu( dot(h[src], tanh(h[dst])) )
//   e_soft = edge_softmax(e, dst)
//   rst    = segment_sum(h[src], dst) * in_deg^-1/2 + bias
// ---------------------------------------------------------------------------

constexpr int N_NODES   = 50000;
constexpr int N_EDGES   = 800000;
constexpr int IN_FEATS  = 256;
constexpr int OUT_FEATS = 128;
constexpr float NEG_SLOPE = 0.2f;

typedef __attribute__((ext_vector_type(2))) float v2f;
typedef __attribute__((ext_vector_type(8))) float v8f;

// --- monotonic float<->int mapping so segment-max can use integer atomicMax
__device__ __forceinline__ int f2ord(float f) {
    int i = __float_as_int(f);
    return (i >= 0) ? i : (i ^ 0x7fffffff);
}
__device__ __forceinline__ float ord2f(int i) {
    return __int_as_float((i >= 0) ? i : (i ^ 0x7fffffff));
}

// ---------------------------------------------------------------------------
__global__ void init_nodes_kernel(float* od, float* id, float* esum, int* emaxi, int n) {
    int i = blockIdx.x * blockDim.x + threadIdx.x;
    if (i >= n) return;
    od[i]    = 0.0f;
    id[i]    = 0.0f;
    esum[i]  = 0.0f;
    emaxi[i] = INT_MIN;   // below ord-code of -inf
}

__global__ void degree_kernel(const int* __restrict__ src, const int* __restrict__ dst,
                              float* od, float* id, int e) {
    int i = blockIdx.x * blockDim.x + threadIdx.x;
    if (i >= e) return;
    atomicAdd(&od[src[i]], 1.0f);
    atomicAdd(&id[dst[i]], 1.0f);
}

__global__ void deg_rsqrt_kernel(float* od, float* id, int n) {
    int i = blockIdx.x * blockDim.x + threadIdx.x;
    if (i >= n) return;
    od[i] = 1.0f / sqrtf(fmaxf(od[i], 1.0f));   // deg^-1/2, clamped
    id[i] = 1.0f / sqrtf(fmaxf(id[i], 1.0f));
}

// ---------------------------------------------------------------------------
// WMMA GEMM: h[M=50000, 128] = (feat * rsод[row]) @ W[256,128]
// grid.x = 3125 M-tiles; 256 threads = 8 waves; wave w owns N-tile w.
// A (16x4 f32):  lanes 0-15 -> K = k0+{0,1};  lanes 16-31 -> K = k0+{2,3}
// B (4x16 f32):  same K split; N = n0 + (lane&15)
// C/D (16x16):   c[v] = D[m0 + v + 8*(lane>=16)][n0 + (lane&15)]
// ---------------------------------------------------------------------------
__global__ void gemm_wmma_kernel(const float* __restrict__ feat,
                                 const float* __restrict__ W,
                                 const float* __restrict__ rs_od,
                                 float* __restrict__ h) {
    const int m0    = blockIdx.x * 16;
    const int lane  = threadIdx.x & 31;
    const int wave  = threadIdx.x >> 5;
    const int n0    = wave * 16;
    const int lrow  = lane & 15;
    const int khalf = (lane >> 4) << 1;       // 0 for lanes 0-15, 2 for 16-31

    const int   row   = m0 + lrow;
    const float scale = rs_od[row];
    const float* __restrict__ arow = feat + (size_t)row * IN_FEATS;
    const int col = n0 + lrow;

    v8f c = {};
    for (int k0 = 0; k0 < IN_FEATS; k0 += 4) {
        v2f a, b;
        a.x = arow[k0 + khalf + 0] * scale;
        a.y = arow[k0 + khalf + 1] * scale;
        b.x = W[(k0 + khalf + 0) * OUT_FEATS + col];
        b.y = W[(k0 + khalf + 1) * OUT_FEATS + col];
        c = __builtin_amdgcn_wmma_f32_16x16x4_f32(
                /*neg_a=*/false, a, /*neg_b=*/false, b,
                /*c_mod=*/(short)0, c, /*reuse_a=*/false, /*reuse_b=*/false);
    }

    const int rowBase = m0 + ((lane >> 4) << 3);   // +8 for upper half-wave
#pragma unroll
    for (int v = 0; v < 8; ++v)
        h[(size_t)(rowBase + v) * OUT_FEATS + col] = c[v];
}

// ---------------------------------------------------------------------------
// Wave-per-edge: attention score + scatter-sum of h[src] into rst (fused).
// Each lane owns a float4 slice of the 128-dim feature vector.
// ---------------------------------------------------------------------------
__global__ void edge_score_scatter_kernel(const float* __restrict__ h,
                                          const int* __restrict__ src,
                                          const int* __restrict__ dst,
                                          float* __restrict__ e,
                                          int* __restrict__ emaxi,
                                          float* __restrict__ rst,
                                          int ne) {
    const int wid  = blockIdx.x * (blockDim.x >> 5) + (threadIdx.x >> 5);
    if (wid >= ne) return;                 // wave-uniform exit
    const int lane = threadIdx.x & 31;
    const int s = src[wid];
    const int d = dst[wid];

    const float4 a = ((const float4*)(h + (size_t)s * OUT_FEATS))[lane];
    const float4 b = ((const float4*)(h + (size_t)d * OUT_FEATS))[lane];

    // scatter-sum (update_all copy_u/sum) — independent of softmax chain
    float* out = rst + (size_t)d * OUT_FEATS + lane * 4;
    atomicAdd(out + 0, a.x);
    atomicAdd(out + 1, a.y);
    atomicAdd(out + 2, a.z);
    atomicAdd(out + 3, a.w);

    float acc = a.x * tanhf(b.x) + a.y * tanhf(b.y)
              + a.z * tanhf(b.z) + a.w * tanhf(b.w);
#pragma unroll
    for (int off = 16; off > 0; off >>= 1)
        acc += __shfl_xor(acc, off, 32);

    if (lane == 0) {
        float ev = (acc > 0.0f) ? acc : NEG_SLOPE * acc;   // leaky_relu(0.2)
        e[wid] = ev;
        atomicMax(&emaxi[d], f2ord(ev));
    }
}

__global__ void edge_exp_kernel(const float* __restrict__ e,
                                const int* __restrict__ dst,
                                const int* __restrict__ emaxi,
                                float* __restrict__ esum,
                                float* __restrict__ esoft, int ne) {
    int i = blockIdx.x * blockDim.x + threadIdx.x;
    if (i >= ne) return;
    int d = dst[i];
    float ee = expf(e[i] - ord2f(emaxi[d]));
    esoft[i] = ee;
    atomicAdd(&esum[d], ee);
}

__global__ void edge_norm_kernel(float* __restrict__ esoft,
                                 const int* __restrict__ dst,
                                 const float* __restrict__ esum, int ne) {
    int i = blockIdx.x * blockDim.x + threadIdx.x;
    if (i >= ne) return;
    esoft[i] = esoft[i] / esum[dst[i]];
}

__global__ void finalize_kernel(float* __restrict__ rst,
                                const float* __restrict__ rs_id,
                                const float* __restrict__ bias, int total) {
    int i = blockIdx.x * blockDim.x + threadIdx.x;
    if (i >= total) return;
    int n = i >> 7;          // / OUT_FEATS
    int k = i & 127;         // % OUT_FEATS
    rst[i] = rst[i] * rs_id[n] + bias[k];
}

// ---------------------------------------------------------------------------
extern "C" void kernel_launch(void* const* d_in, const int* in_sizes, int n_in,
                              void* d_out, int out_size, void* d_ws, size_t ws_size,
                              hipStream_t stream) {
    const float* feat = (const float*)d_in[0];   // [N, 256]
    const float* W    = (const float*)d_in[1];   // [256, 128]
    const float* bias = (const float*)d_in[2];   // [128]
    const int*   src  = (const int*)  d_in[3];   // [E]
    const int*   dst  = (const int*)  d_in[4];   // [E]

    float* rst   = (float*)d_out;                          // [N, 128]
    float* esoft = rst + (size_t)N_NODES * OUT_FEATS;      // [E]

    // Workspace carve-up (~29.6 MB)
    char* p = (char*)d_ws;
    float* h     = (float*)p; p += sizeof(float) * (size_t)N_NODES * OUT_FEATS;
    float* od    = (float*)p; p += sizeof(float) * N_NODES;   // becomes out_deg^-1/2
    float* id    = (float*)p; p += sizeof(float) * N_NODES;   // becomes in_deg^-1/2
    float* e     = (float*)p; p += sizeof(float) * N_EDGES;
    int*   emaxi = (int*)  p; p += sizeof(int)   * N_NODES;
    float* esum  = (float*)p; p += sizeof(float) * N_NODES;

    const int T = 256;
    const int nBlk = (N_NODES + T - 1) / T;          // 196
    const int eBlk = (N_EDGES + T - 1) / T;          // 3125
    const int wBlk = (N_EDGES + 7) / 8;              // 100000 (8 waves/block)
    const int fTot = N_NODES * OUT_FEATS;
    const int fBlk = (fTot + T - 1) / T;             // 25000

    // zero accumulation target for rst (deterministic per call)
    hipMemsetAsync(d_out, 0, sizeof(float) * (size_t)fTot, stream);

    init_nodes_kernel<<<nBlk, T, 0, stream>>>(od, id, esum, emaxi, N_NODES);
    degree_kernel<<<eBlk, T, 0, stream>>>(src, dst, od, id, N_EDGES);
    deg_rsqrt_kernel<<<nBlk, T, 0, stream>>>(od, id, N_NODES);

    gemm_wmma_kernel<<<N_NODES / 16, 256, 0, stream>>>(feat, W, od, h);

    edge_score_scatter_kernel<<<wBlk, 256, 0, stream>>>(h, src, dst, e, emaxi, rst, N_EDGES);
    edge_exp_kernel<<<eBlk, T, 0, stream>>>(e, dst, emaxi, esum, esoft, N_EDGES);
    edge_norm_kernel<<<eBlk, T, 0, stream>>>(esoft, dst, esum, N_EDGES);

    finalize_kernel<<<fBlk, T, 0, stream>>>(rst, id, bias, fTot);
}